// TensorProductNaive_75548474737352
// MI455X (gfx1250) — compile-verified
//
#include <hip/hip_runtime.h>
#include <stdint.h>

// ======================================================================
// MI455X (gfx1250) e3nn tensor product  (128x0e+64x1o+32x2e) x (1x0e+1x1o+1x2e)
// Bandwidth-bound (~315 MB @ 23.3 TB/s => ~13.5us floor).  Strategy:
//   - TDM (tensor_load_to_lds) async-stages x1/x2/CG tiles into LDS
//   - per-edge fold W[e,r] = sum_j b[e,j]*CGexp[j,r] done with
//     V_WMMA_F32_16X16X4_F32 (M=16 edges, K=9->12, N=NW tiled by 16)
//   - per-(edge,u): out[k] = sum_i a[i]*W[i,k], coalesced NT stores
// CG coefficients computed on-device in double (Racah + real-basis change).
// ======================================================================

typedef unsigned int u32x4 __attribute__((ext_vector_type(4)));
typedef int          i32x8 __attribute__((ext_vector_type(8)));
typedef int          i32x4 __attribute__((ext_vector_type(4)));
typedef float        v2f   __attribute__((ext_vector_type(2)));
typedef float        v8f   __attribute__((ext_vector_type(8)));

// ---- chunk metadata: 19 (l1,l2,lo) paths in reference generation order ----
__constant__ int d_CHL1[19] = {0,0,0,1,1,1,1,1,1,1,2,2,2,2,2,2,2,2,2};
__constant__ int d_CHL2[19] = {0,1,2,0,1,1,1,2,2,2,0,1,1,1,2,2,2,2,2};
__constant__ int d_CHLO[19] = {0,1,2,1,0,1,2,1,2,3,2,1,2,3,0,1,2,3,4};
__constant__ int d_CGOFF[19]= {0,1,10,35,44,53,80,125,170,245,350,375,420,495,600,625,700,825,1000};
// ws float layout: [0,1225) dense CG; CGexp matrices below:
#define EXPO_G0 1232
#define EXPO_G1 1376
#define EXPO_G2 2240
// total ws floats = 2240 + 9*240 = 4400  (17.6 KB)

// ----------------------------------------------------------------------
// TDM: 2-D tile descriptor (data_size = 4 bytes), 6-arg builtin form.
// ----------------------------------------------------------------------
__device__ __forceinline__ void tdm_load_2d(uint32_t lds_off, const void* gptr,
    uint32_t tile_d0, uint32_t tile_d1, uint32_t tens_d0, uint32_t tens_d1,
    uint64_t stride0_elems)
{
  uint64_t ga = (uint64_t)(uintptr_t)gptr;
  u32x4 g0;
  g0.x = 1u;                                                 // count=1, user mode
  g0.y = lds_off;                                            // LDS byte address
  g0.z = (uint32_t)ga;                                       // global addr [31:0]
  g0.w = (uint32_t)((ga >> 32) & 0x01FFFFFFu) | (2u << 30);  // addr[56:32], type=2
  i32x8 g1;
  g1[0] = (int)(2u << 16);                                   // data_size = 4 bytes
  g1[1] = (int)((tens_d0 & 0xFFFFu) << 16);                  // tensor_dim0[15:0]
  g1[2] = (int)(((tens_d0 >> 16) & 0xFFFFu) | ((tens_d1 & 0xFFFFu) << 16));
  g1[3] = (int)(((tens_d1 >> 16) & 0xFFFFu) | ((tile_d0 & 0xFFFFu) << 16));
  g1[4] = (int)(tile_d1 & 0xFFFFu);                          // tile_dim1, tile_dim2=0
  g1[5] = (int)(uint32_t)(stride0_elems & 0xFFFFFFFFu);      // dim0 stride [31:0]
  g1[6] = (int)(uint32_t)((stride0_elems >> 32) & 0xFFFFu);  // dim0 stride [47:32]
  g1[7] = 0;
  i32x4 g2 = {0,0,0,0};
  i32x4 g3 = {0,0,0,0};
  i32x8 g4 = {0,0,0,0,0,0,0,0};
  __builtin_amdgcn_tensor_load_to_lds(g0, g1, g2, g3, g4, 0);
}

// ----------------------------------------------------------------------
// On-device Clebsch-Gordan (real basis, e3nn convention, unit Frobenius norm)
// ----------------------------------------------------------------------
__device__ double dfact(int n){ double r=1.0; for(int i=2;i<=n;++i) r*=(double)i; return r; }

__device__ double cg_su2(int j1,int m1,int j2,int m2,int j3,int m3){
  if (m3 != m1+m2) return 0.0;
  double pref = (double)(2*j3+1) * dfact(j1+j2-j3)*dfact(j1-j2+j3)*dfact(-j1+j2+j3)/dfact(j1+j2+j3+1);
  pref *= dfact(j3+m3)*dfact(j3-m3)*dfact(j1-m1)*dfact(j1+m1)*dfact(j2-m2)*dfact(j2+m2);
  int kmin = 0;
  if (j2-j3-m1 > kmin) kmin = j2-j3-m1;
  if (j1-j3+m2 > kmin) kmin = j1-j3+m2;
  int kmax = j1+j2-j3;
  if (j1-m1 < kmax) kmax = j1-m1;
  if (j2+m2 < kmax) kmax = j2+m2;
  double s = 0.0;
  for (int k=kmin;k<=kmax;++k){
    double d = dfact(k)*dfact(j1+j2-j3-k)*dfact(j1-m1-k)*dfact(j2+m2-k)
             * dfact(j3-j2+m1+k)*dfact(j3-j1-m2+k);
    s += ((k & 1) ? -1.0 : 1.0) / d;
  }
  return sqrt(pref) * s;
}

__device__ void build_q(int l, double* qr, double* qi){
  int n = 2*l+1;
  for (int x=0;x<n*n;++x){ qr[x]=0.0; qi[x]=0.0; }
  const double s2 = 0.70710678118654752440;
  for (int m=-l;m<0;++m){
    int am = -m;
    qr[(l+m)*n + (l+am)] =  s2;     //  1/sqrt2
    qi[(l+m)*n + (l-am)] = -s2;     // -i/sqrt2
  }
  qr[l*n + l] = 1.0;
  for (int m=1;m<=l;++m){
    double sgn = (m & 1) ? -1.0 : 1.0;
    qr[(l+m)*n + (l+m)] = sgn * s2;
    qi[(l+m)*n + (l-m)] = sgn * s2; // i*(-1)^m/sqrt2
  }
  int ph = l & 3;                    // multiply everything by (-i)^l
  if (ph){
    for (int x=0;x<n*n;++x){
      double re=qr[x], im=qi[x];
      if      (ph==1){ qr[x]= im; qi[x]=-re; }  // * -i
      else if (ph==2){ qr[x]=-re; qi[x]=-im; }  // * -1
      else           { qr[x]=-im; qi[x]= re; }  // *  i
    }
  }
}

__global__ void cg_prep_kernel(float* __restrict__ cgbuf){
  int t = threadIdx.x;
  if (t >= 19) return;
  int l1=d_CHL1[t], l2=d_CHL2[t], l3=d_CHLO[t];
  int n1=2*l1+1, n2=2*l2+1, n3=2*l3+1;
  double q1r[81],q1i[81],q2r[81],q2i[81],q3r[81],q3i[81];
  build_q(l1,q1r,q1i); build_q(l2,q2r,q2i); build_q(l3,q3r,q3i);
  double R[225]; double nrm2 = 0.0;
  for (int a=0;a<n1;++a)
  for (int b=0;b<n2;++b)
  for (int c=0;c<n3;++c){
    double acc = 0.0;
    for (int mi1=0;mi1<n1;++mi1)
    for (int mi2=0;mi2<n2;++mi2){
      int m3 = (mi1-l1)+(mi2-l2);
      if (m3 < -l3 || m3 > l3) continue;
      double w = cg_su2(l1, mi1-l1, l2, mi2-l2, l3, m3);
      if (w == 0.0) continue;
      int mi3 = m3 + l3;
      double ar=q1r[mi1*n1+a], ai=q1i[mi1*n1+a];
      double br=q2r[mi2*n2+b], bi=q2i[mi2*n2+b];
      double cr=q3r[mi3*n3+c], ci=q3i[mi3*n3+c];
      double pr = ar*br - ai*bi;
      double pi = ar*bi + ai*br;
      acc += w * (pr*cr + pi*ci);   // Re(Q1*Q2*conj(Q3))
    }
    R[(a*n2+b)*n3+c] = acc;
    nrm2 += acc*acc;
  }
  double inv = 1.0 / sqrt(nrm2);
  float* o = cgbuf + d_CGOFF[t];
  int sz = n1*n2*n3;
  for (int x=0;x<sz;++x) o[x] = (float)(R[x]*inv);
}

// ----------------------------------------------------------------------
// Per-l1 group parameters.  Output column bases derive from e3nn's stable
// sort: 0e[0) 1o[224) 1e[1088) 2e[1376) 2o[2656) 3o[3136) 3e[3808) 4e[4032).
// EPB=16 edges per block (one WMMA M-tile), 512 threads = 16 waves.
// ----------------------------------------------------------------------
template<int G> struct GP;
template<> struct GP<0>{
  static constexpr int MUL=128, NA=1, XOFF=0,   XW=128, NW=9,   NWP=16,  EXPO=EXPO_G0, NCH=3;
  static constexpr int CGO[NCH]={0,1,10};
  static constexpr int NJ [NCH]={1,3,5};
  static constexpr int JO [NCH]={0,1,4};
  static constexpr int NK [NCH]={1,3,5};
  static constexpr int OB [NCH]={0,224,1376};
};
template<> struct GP<1>{
  static constexpr int MUL=64,  NA=3, XOFF=128, XW=192, NW=81,  NWP=96,  EXPO=EXPO_G1, NCH=7;
  static constexpr int CGO[NCH]={35,44,53,80,125,170,245};
  static constexpr int NJ [NCH]={1,3,3,3,5,5,5};
  static constexpr int JO [NCH]={0,1,1,1,4,4,4};
  static constexpr int NK [NCH]={3,1,3,5,3,5,7};
  static constexpr int OB [NCH]={608,128,1088,2016,800,2656,3136};
};
template<> struct GP<2>{
  static constexpr int MUL=32,  NA=5, XOFF=320, XW=160, NW=225, NWP=240, EXPO=EXPO_G2, NCH=9;
  static constexpr int CGO[NCH]={350,375,420,495,600,625,700,825,1000};
  static constexpr int NJ [NCH]={1,3,3,3,5,5,5,5,5};
  static constexpr int JO [NCH]={0,1,1,1,4,4,4,4,4};
  static constexpr int NK [NCH]={5,3,5,7,1,3,5,7,9};
  static constexpr int OB [NCH]={2336,992,2976,3584,192,1280,2496,3808,4032};
};

// Expand dense CG into the WMMA B-matrix  CGexp[j, r]  (9 x NWP, zero-padded):
// r decodes to (chunk c, i, k); column value = cg[(i*NJ + (j-JO))*NK + k].
template<int G>
__global__ void cg_expand_kernel(const float* __restrict__ cg, float* __restrict__ ws){
  using P = GP<G>;
  int idx = blockIdx.x*256 + threadIdx.x;
  if (idx >= 9*P::NWP) return;
  int j = idx / P::NWP, r = idx - j*P::NWP;
  float v = 0.f;
  int wb = 0;
#pragma unroll
  for (int c=0;c<P::NCH;++c){
    const int sz = P::NA*P::NK[c];
    if (r >= wb && r < wb+sz){
      int rr = r - wb;
      int i = rr / P::NK[c], k = rr - i*P::NK[c];
      int jl = j - P::JO[c];
      if (jl >= 0 && jl < P::NJ[c])
        v = cg[P::CGO[c] + (i*P::NJ[c]+jl)*P::NK[c] + k];
    }
    wb += sz;
  }
  ws[P::EXPO + idx] = v;
}

template<int G>
__global__ __launch_bounds__(512) void tp_kernel(
    const float* __restrict__ x1, const float* __restrict__ x2,
    const float* __restrict__ wsf, float* __restrict__ out, int n_edges)
{
  using P = GP<G>;
  constexpr int MUL=P::MUL, NA=P::NA, XW=P::XW, NW=P::NW, NWP=P::NWP, NCH=P::NCH;
  constexpr int EPB = 16;                 // one WMMA M-tile of edges
  constexpr int REP = (EPB*MUL)/512;      // (edge,u) pairs per thread
  constexpr int NTILES = NWP/16;
  __shared__ float sA [EPB*XW];           // x1 slab for this l1 group
  __shared__ float sB [EPB*9];            // x2 rows (WMMA A operand source)
  __shared__ float sCG[9*NWP];            // CGexp   (WMMA B operand source)
  __shared__ float sW [EPB*NWP];          // folded W per edge (WMMA D)

  const int tid   = threadIdx.x;
  const int edge0 = blockIdx.x * EPB;
  int validE = n_edges - edge0; if (validE > EPB) validE = EPB;

  // ---- wave 0 issues async TDM loads; all wait at the barrier ----
  if (tid < 32) {
    tdm_load_2d((uint32_t)(uintptr_t)&sA[0],
                x1 + (size_t)edge0*480 + P::XOFF,
                (uint32_t)XW, (uint32_t)validE,
                (uint32_t)XW, (uint32_t)validE, 480ull);
    tdm_load_2d((uint32_t)(uintptr_t)&sB[0],
                x2 + (size_t)edge0*9,
                (uint32_t)(9*validE), 1u,
                (uint32_t)(9*validE), 1u, (uint64_t)(9*validE));
    tdm_load_2d((uint32_t)(uintptr_t)&sCG[0],
                wsf + P::EXPO,
                (uint32_t)(9*NWP), 1u,
                (uint32_t)(9*NWP), 1u, (uint64_t)(9*NWP));
    __builtin_amdgcn_s_wait_tensorcnt(0);
  }
  __syncthreads();

  // ---- WMMA fold: W(16 edges x NWP) = b(16x9,pad12) * CGexp(9x NWP,pad12) ----
  // A-frag (16x4 f32): lane 0-15 row M, VGPR0=K(koff), VGPR1=K(koff+1), koff=2*(lane/16)
  // B-frag (4x16 f32): same K striping, col = ntile*16 + (lane&15)
  // D-frag (16x16 f32): VGPR r -> row (r | r+8), col = lane&15 (+0/+16 half)
  {
    const int wv   = tid >> 5;
    const int lane = tid & 31;
    const int half = lane >> 4;           // 0: lanes 0-15, 1: lanes 16-31
    const int koff = half << 1;
    for (int nt = wv; nt < NTILES; nt += 16) {
      const int col = nt*16 + (lane & 15);
      const int e   = lane & 15;
      v8f acc = {0.f,0.f,0.f,0.f,0.f,0.f,0.f,0.f};
#pragma unroll
      for (int s = 0; s < 3; ++s) {       // K = 9 padded to 12
        const int j0 = 4*s + koff, j1 = j0 + 1;
        v2f av, bv;
        av.x = (j0 < 9) ? sB[e*9 + j0]      : 0.f;
        av.y = (j1 < 9) ? sB[e*9 + j1]      : 0.f;
        bv.x = (j0 < 9) ? sCG[j0*NWP + col] : 0.f;
        bv.y = (j1 < 9) ? sCG[j1*NWP + col] : 0.f;
        acc = __builtin_amdgcn_wmma_f32_16x16x4_f32(
                  false, av, false, bv, (short)0, acc, false, false);
      }
#pragma unroll
      for (int r = 0; r < 8; ++r)
        sW[(r + (half<<3))*NWP + col] = acc[r];
    }
  }
  __syncthreads();

  // ---- per-(edge,u): out[k] = sum_i a[i]*W[i,k]; coalesced NT stores ----
#pragma unroll
  for (int rep = 0; rep < REP; ++rep) {
    const int p = tid + rep*512;
    const int e = p / MUL, u = p - e*MUL;
    if (e < validE) {
      float a[NA];
#pragma unroll
      for (int i=0;i<NA;++i) a[i] = sA[e*XW + u*NA + i];
      float* orow = out + (size_t)(edge0 + e) * 4320;
      const float* we = sW + e*NWP;
      int wb = 0;
#pragma unroll
      for (int c=0;c<NCH;++c){
#pragma unroll
        for (int k=0;k<P::NK[c];++k){
          float s = 0.f;
#pragma unroll
          for (int i=0;i<NA;++i)
            s = fmaf(a[i], we[wb + i*P::NK[c] + k], s);
          __builtin_nontemporal_store(s, orow + P::OB[c] + u*P::NK[c] + k);
        }
        wb += NA*P::NK[c];
      }
    }
  }
}

// ----------------------------------------------------------------------
extern "C" void kernel_launch(void* const* d_in, const int* in_sizes, int n_in,
                              void* d_out, int out_size, void* d_ws, size_t ws_size,
                              hipStream_t stream)
{
  const float* x1   = (const float*)d_in[0];
  const float* x2   = (const float*)d_in[1];
  float*       out  = (float*)d_out;
  float*       wsf  = (float*)d_ws;        // CG tables + CGexp matrices
  const int n_edges = in_sizes[0] / 480;

  cg_prep_kernel<<<1, 32, 0, stream>>>(wsf);
  cg_expand_kernel<0><<<(9*GP<0>::NWP + 255)/256, 256, 0, stream>>>(wsf, wsf);
  cg_expand_kernel<1><<<(9*GP<1>::NWP + 255)/256, 256, 0, stream>>>(wsf, wsf);
  cg_expand_kernel<2><<<(9*GP<2>::NWP + 255)/256, 256, 0, stream>>>(wsf, wsf);

  const int blocks = (n_edges + 15)/16;
  tp_kernel<0><<<blocks, 512, 0, stream>>>(x1, x2, wsf, out, n_edges);
  tp_kernel<1><<<blocks, 512, 0, stream>>>(x1, x2, wsf, out, n_edges);
  tp_kernel<2><<<blocks, 512, 0, stream>>>(x1, x2, wsf, out, n_edges);
  (void)n_in; (void)out_size; (void)ws_size;
}